// CopyNet_11012296147232
// MI455X (gfx1250) — compile-verified
//
#include <hip/hip_runtime.h>

// CopyNet pointer-generator mixture for MI455X (gfx1250).
//   out[b,t,v] = p_gen[b,t] * dist[b,t,v]  (v < V; 0 for padded v)
//              + (1 - p_gen[b,t]) * sum_{s: ptr[b,s]==v} alpha[b,s,t]
//
// Pass 1: bandwidth-bound stream (262 MB read + 263 MB write), NT cache hints
//         (one-touch data, working set > 192 MB L2).
// Pass 2: 1M native fp32 global atomics (scatter-add), negligible traffic.

#define NB      8
#define NLDEC   256
#define NV      32000
#define NLSRC   512
#define NVEXT   32128

#define NV4     (NV / 4)      // 8000 float4 per dist row
#define NVEXT4  (NVEXT / 4)   // 8032 float4 per out row

// clang-native 128-bit vector (accepted by __builtin_nontemporal_*)
typedef float v4f __attribute__((ext_vector_type(4)));

__global__ __launch_bounds__(256) void copynet_vocab_scale_kernel(
    const float* __restrict__ dist,   // [NB*NLDEC, NV]
    const float* __restrict__ pgen,   // [NB*NLDEC]
    float* __restrict__ out)          // [NB*NLDEC, NVEXT]
{
    const int r  = blockIdx.y;                       // row = b*NLDEC + t
    const int c4 = blockIdx.x * 256 + threadIdx.x;   // float4 column
    if (c4 >= NVEXT4) return;

    v4f* __restrict__ orow = (v4f*)(out + (size_t)r * NVEXT);

    if (c4 < NV4) {
        const v4f* __restrict__ drow = (const v4f*)(dist + (size_t)r * NV);
        const float pg = pgen[r];                    // uniform -> s_load
        v4f d = __builtin_nontemporal_load(&drow[c4]);   // b128, TH_NT
        d *= pg;
        __builtin_nontemporal_store(d, &orow[c4]);       // b128, TH_NT
    } else {
        // padded vocab slots [NV, NVEXT): base probability is zero
        v4f z = {0.0f, 0.0f, 0.0f, 0.0f};
        __builtin_nontemporal_store(z, &orow[c4]);
    }
}

__global__ __launch_bounds__(256) void copynet_scatter_kernel(
    const float* __restrict__ alph,     // [NB, NLSRC, NLDEC]
    const float* __restrict__ pgen,     // [NB*NLDEC]
    const int*   __restrict__ pointer,  // [NB, NLSRC]
    float* __restrict__ out)            // [NB*NLDEC, NVEXT]
{
    const int b = blockIdx.y;     // batch
    const int s = blockIdx.x;     // source position
    const int t = threadIdx.x;    // decoder step (256 = 8 waves)

    const int v = pointer[b * NLSRC + s];                        // uniform scalar
    const float a = alph[((size_t)(b * NLSRC + s)) * NLDEC + t]; // coalesced
    const float w = 1.0f - pgen[b * NLDEC + t];                  // coalesced

    float* addr = out + ((size_t)(b * NLDEC + t)) * NVEXT + (size_t)v;
    // relaxed agent-scope fp32 add -> native global_atomic_add_f32
    __hip_atomic_fetch_add(addr, w * a, __ATOMIC_RELAXED,
                           __HIP_MEMORY_SCOPE_AGENT);
}

extern "C" void kernel_launch(void* const* d_in, const int* in_sizes, int n_in,
                              void* d_out, int out_size, void* d_ws, size_t ws_size,
                              hipStream_t stream) {
    const float* dist    = (const float*)d_in[0];   // [8,256,32000] f32
    const float* pgen    = (const float*)d_in[1];   // [8,256,1]     f32
    const float* alph    = (const float*)d_in[2];   // [8,512,256]   f32
    // d_in[3] = batch_vocab: only its shape (V_EXT) matters -> compile-time const
    const int*   pointer = (const int*)d_in[4];     // [8,512]       i32
    float* out = (float*)d_out;                     // [8,256,32128] f32

    // Pass 1: out = p_gen * [dist ; 0-pad]   (full overwrite of d_out)
    dim3 grid1((NVEXT4 + 255) / 256, NB * NLDEC);   // (32, 2048)
    copynet_vocab_scale_kernel<<<grid1, 256, 0, stream>>>(dist, pgen, out);

    // Pass 2: out[b,t,ptr[b,s]] += (1 - p_gen[b,t]) * alpha[b,s,t]
    dim3 grid2(NLSRC, NB);                          // (512, 8)
    copynet_scatter_kernel<<<grid2, NLDEC, 0, stream>>>(alph, pgen, pointer, out);
}